// EncdecMaskPadItemLoss_8486855377000
// MI455X (gfx1250) — compile-verified
//
#include <hip/hip_runtime.h>
#include <cstdint>
#include <cstddef>

#define B_DIM 8
#define S_DIM 1024
#define V_DIM 32000
#define NROWS (B_DIM * S_DIM)
#define BLK 256
#define CHUNK 6400               // floats per chunk -> 25.6 KB, 5 chunks per row
#define NCHUNK (V_DIM / CHUNK)   // 5
#define CHUNK_B128 (CHUNK / 4)   // 1600 16-byte async copies / float4s per chunk

static_assert(CHUNK * NCHUNK == V_DIM, "chunking must tile V exactly");
static_assert((CHUNK * 4) % 16 == 0, "buffers must stay 16B aligned");

// Low 32 bits of a generic pointer into __shared__ = LDS byte offset
// (hardware adds the wave's LDS_BASE; async-load VDST VGPR wants this offset).
__device__ __forceinline__ unsigned lds_offset_of(const void* p) {
  return (unsigned)(uintptr_t)p;
}

__device__ __forceinline__ void async_copy_chunk_to_lds(const float* __restrict__ gsrc,
                                                        const float* lds_dst,
                                                        int tid) {
  // strength-reduced: per-iteration address update is a constant add
  unsigned laddr = lds_offset_of(lds_dst) + (unsigned)tid * 16u;
  const float* g = gsrc + tid * 4;
  for (int c = tid; c < CHUNK_B128; c += BLK) {
    // CDNA5 async global->LDS DMA, tracked with ASYNCcnt (no VGPR data return)
    asm volatile("global_load_async_to_lds_b128 %0, %1, off"
                 :: "v"(laddr), "v"(g)
                 : "memory");
    laddr += BLK * 16u;
    g += BLK * 4;
  }
}

__global__ void __launch_bounds__(BLK)
row_logsumexp_nll_kernel(const float* __restrict__ logits,
                         const int* __restrict__ tgt,
                         float* __restrict__ nll_out) {
  __shared__ float buf[2][CHUNK];   // double-buffered streaming tile (51.2 KB)
  __shared__ float mred[BLK];
  __shared__ float sred[BLK];

  const int row = blockIdx.x;
  const int tid = threadIdx.x;
  const float* __restrict__ gbase = logits + (size_t)row * V_DIM;

  // prologue: kick chunk 0 into buf[0]
  async_copy_chunk_to_lds(gbase, &buf[0][0], tid);

  float m_t = -3.402823466e38f;  // per-thread running max
  float s_t = 0.0f;              // per-thread running sum of exp(x - m_t)

  for (int k = 0; k < NCHUNK; ++k) {
    // drain this wave's outstanding async copies (chunk k), then block barrier
    // so every wave's slice of the tile is visible in LDS.
    asm volatile("s_wait_asynccnt 0x0" ::: "memory");
    __syncthreads();

    // overlap: start streaming chunk k+1 while we reduce chunk k.
    // Safe: all threads passed the barrier above, so nobody is still reading
    // buf[(k+1)&1] (last touched as chunk k-1).
    if (k + 1 < NCHUNK) {
      async_copy_chunk_to_lds(gbase + (size_t)(k + 1) * CHUNK,
                              &buf[(k + 1) & 1][0], tid);
    }

    // wide LDS reads: ds_load_b128 per 4 elements
    const float4* __restrict__ bp4 =
        reinterpret_cast<const float4*>(&buf[k & 1][0]);

    // pass 1 (VALU only): chunk-local max over this thread's strided slice
    float cm = -3.402823466e38f;
    for (int i = tid; i < CHUNK_B128; i += BLK) {
      float4 v = bp4[i];
      cm = fmaxf(cm, fmaxf(fmaxf(v.x, v.y), fmaxf(v.z, v.w)));
    }
    float nm = fmaxf(m_t, cm);
    // rescale running sum once per chunk, then one exp per element
    s_t *= __expf(m_t - nm);
    float acc = 0.0f;
    for (int i = tid; i < CHUNK_B128; i += BLK) {
      float4 v = bp4[i];
      acc += __expf(v.x - nm) + __expf(v.y - nm) +
             __expf(v.z - nm) + __expf(v.w - nm);
    }
    s_t += acc;
    m_t = nm;
  }

  // merge 256 per-thread (m, s) pairs: logsumexp pair-combine tree
  mred[tid] = m_t;
  sred[tid] = s_t;
  __syncthreads();
  for (int off = BLK / 2; off > 0; off >>= 1) {
    if (tid < off) {
      float m1 = mred[tid], m2 = mred[tid + off];
      float s1 = sred[tid], s2 = sred[tid + off];
      float mm = fmaxf(m1, m2);
      sred[tid] = s1 * __expf(m1 - mm) + s2 * __expf(m2 - mm);
      mred[tid] = mm;
    }
    __syncthreads();
  }
  if (tid == 0) {
    int t = tgt[row];
    float xt = gbase[t];  // single scalar gather straight from global
    nll_out[row] = (mred[0] + __logf(sred[0])) - xt;
  }
}

__global__ void __launch_bounds__(BLK)
finalize_kernel(const float* __restrict__ nll,
                const int* __restrict__ tok,
                float* __restrict__ out) {
  __shared__ float rsum[3][BLK];
  __shared__ float rcnt[3][BLK];
  __shared__ float mean_[3][B_DIM];
  __shared__ float pres_[3][B_DIM];
  const int tid = threadIdx.x;

  for (int b = 0; b < B_DIM; ++b) {
    float s[3] = {0.0f, 0.0f, 0.0f};
    float c[3] = {0.0f, 0.0f, 0.0f};
    for (int i = tid; i < S_DIM; i += BLK) {
      int t = tok[b * S_DIM + i];
      float v = nll[b * S_DIM + i];
      // cat 0 = regular, 1 = mask token (==4), 2 = special (0..3)
      int cat = (t == 4) ? 1 : (((unsigned)t <= 3u) ? 2 : 0);
      s[cat] += v;
      c[cat] += 1.0f;
    }
    for (int k = 0; k < 3; ++k) { rsum[k][tid] = s[k]; rcnt[k][tid] = c[k]; }
    __syncthreads();
    for (int off = BLK / 2; off > 0; off >>= 1) {
      if (tid < off) {
        for (int k = 0; k < 3; ++k) {
          rsum[k][tid] += rsum[k][tid + off];
          rcnt[k][tid] += rcnt[k][tid + off];
        }
      }
      __syncthreads();
    }
    if (tid == 0) {
      for (int k = 0; k < 3; ++k) {
        float cc = rcnt[k][0];
        pres_[k][b] = (cc > 0.0f) ? 1.0f : 0.0f;
        mean_[k][b] = (cc > 0.0f) ? rsum[k][0] / cc : 0.0f;
      }
    }
    __syncthreads();
  }

  if (tid == 0) {
    const float W[3] = {1.0f, 1.0f, 0.01f};  // REG_W, MSK_W, SPC_W
    float loss = 0.0f;
    float avg[3] = {0.0f, 0.0f, 0.0f};
    float npres[3] = {0.0f, 0.0f, 0.0f};
    for (int b = 0; b < B_DIM; ++b) {
      float num = 0.0f, tw = 0.0f;
      for (int k = 0; k < 3; ++k) {
        num += mean_[k][b] * W[k] * pres_[k][b];
        tw  += pres_[k][b] * W[k];
      }
      loss += num / tw;
      for (int k = 0; k < 3; ++k) {
        avg[k] += mean_[k][b];
        npres[k] += pres_[k][b];
      }
    }
    out[0] = loss / (float)B_DIM;
    for (int k = 0; k < 3; ++k) {
      out[1 + k] = (npres[k] > 0.0f) ? avg[k] / fmaxf(npres[k], 1.0f) : 0.0f;
    }
  }
}

extern "C" void kernel_launch(void* const* d_in, const int* in_sizes, int n_in,
                              void* d_out, int out_size, void* d_ws, size_t ws_size,
                              hipStream_t stream) {
  (void)in_sizes; (void)n_in; (void)out_size; (void)ws_size;
  const float* logits = (const float*)d_in[0];  // [8,1024,32000] f32
  const int* tok_inp  = (const int*)d_in[1];    // [8,1024] i32
  const int* tok_tgt  = (const int*)d_in[2];    // [8,1024] i32
  float* out = (float*)d_out;                   // 4 scalars
  float* nll = (float*)d_ws;                    // [8192] f32 scratch (fully overwritten)

  row_logsumexp_nll_kernel<<<NROWS, BLK, 0, stream>>>(logits, tok_tgt, nll);
  finalize_kernel<<<1, BLK, 0, stream>>>(nll, tok_inp, out);
}